// TemporalAggregator_23407571763656
// MI455X (gfx1250) — compile-verified
//
#include <hip/hip_runtime.h>
#include <math.h>

typedef __attribute__((ext_vector_type(2))) float v2f;
typedef __attribute__((ext_vector_type(4))) float v4f;
typedef __attribute__((ext_vector_type(8))) float v8f;

#define BB 16
#define SS 512
#define TT 64
#define FF 256
#define GROUP 16      // (b,s) pairs per block
#define THREADS 256   // 8 waves of 32

// wave32 butterfly add via ds_swizzle group-of-32 XOR patterns
// offset = (xor_mask<<10) | (or_mask<<5) | and_mask ; and_mask=0x1f
template <int PATTERN>
__device__ __forceinline__ float swizzle_add(float v) {
  int j = __builtin_amdgcn_ds_swizzle(__float_as_int(v), PATTERN);
  return v + __int_as_float(j);
}

__device__ __forceinline__ float wave_reduce_add(float v) {
  v = swizzle_add<0x401F>(v);  // SWAPX16
  v = swizzle_add<0x201F>(v);  // SWAPX8
  v = swizzle_add<0x101F>(v);  // SWAPX4
  v = swizzle_add<0x081F>(v);  // SWAPX2
  v = swizzle_add<0x041F>(v);  // SWAPX1
  return v;
}

__global__ __launch_bounds__(THREADS)
void temporal_agg_kernel(const float* __restrict__ x,
                         const float* __restrict__ w,
                         float* __restrict__ out) {
  // LDS: 16 query vectors (Q[p][d]) and the projected queries (WQ[p][f])
  __shared__ __align__(16) float q_lds[GROUP * FF];   // 16 KB
  __shared__ __align__(16) float wq_lds[GROUP * FF];  // 16 KB

  const int tid  = threadIdx.x;
  const int wave = tid >> 5;
  const int lane = tid & 31;
  const long pair0 = (long)blockIdx.x * GROUP;  // 16 consecutive (b*S+s) pairs

  // ---- Phase 0: stage the 16 queries q = x[b,s,T-1,:] into LDS -------------
  for (int i = tid; i < GROUP * FF; i += THREADS) {
    const int p = i >> 8;          // FF == 256
    const int d = i & (FF - 1);
    q_lds[i] = x[(pair0 + p) * (TT * FF) + (TT - 1) * FF + d];
  }
  __syncthreads();

  // ---- Phase 1: WQ[f,p] = sum_d W[f,d] * Q[p,d] via V_WMMA_F32_16X16X4_F32 -
  // A = W (M=f, K=d), B = Q^T (K=d, N=p). 16 M-tiles of 16; wave w owns 2.
  {
    const int m    = lane & 15;          // row within tile (A) / column p (B,D)
    const int koff = (lane >> 4) << 1;   // K half-split across lane halves
    const int f0 = (wave * 2 + 0) * 16;
    const int f1 = (wave * 2 + 1) * 16;
    v8f acc0 = {};
    v8f acc1 = {};
    const float* wrow0 = w + (long)(f0 + m) * FF + koff;
    const float* wrow1 = w + (long)(f1 + m) * FF + koff;
    const float* qrow  = q_lds + m * FF + koff;
#pragma unroll 4
    for (int k = 0; k < FF; k += 4) {
      // A frag: lanes 0-15 -> W[f0+m][k..k+1], lanes 16-31 -> W[f0+m][k+2..k+3]
      v2f a0 = *(const v2f*)(wrow0 + k);
      v2f a1 = *(const v2f*)(wrow1 + k);
      // B frag: lanes 0-15 -> Q[p][k..k+1], lanes 16-31 -> Q[p][k+2..k+3]
      v2f bq = *(const v2f*)(qrow + k);
      acc0 = __builtin_amdgcn_wmma_f32_16x16x4_f32(
          false, a0, false, bq, (short)0, acc0, false, false);
      acc1 = __builtin_amdgcn_wmma_f32_16x16x4_f32(
          false, a1, false, bq, (short)0, acc1, false, false);
    }
    // D layout: VGPR r -> M=r (lanes 0-15) / M=r+8 (lanes 16-31), N=lane&15
    const int p     = lane & 15;
    const int mbase = (lane < 16) ? 0 : 8;
#pragma unroll
    for (int r = 0; r < 8; ++r) {
      wq_lds[p * FF + f0 + mbase + r] = acc0[r];
      wq_lds[p * FF + f1 + mbase + r] = acc1[r];
    }
  }
  __syncthreads();

  // ---- Phase 2: streaming pass over X with online softmax ------------------
  // wave w owns local pairs 2w (A) and 2w+1 (B), processed CONCURRENTLY for
  // memory-level parallelism; lane l owns features [8l, 8l+8).
  {
    const int  pA = wave * 2 + 0;
    const int  pB = wave * 2 + 1;
    const long pairA = pair0 + pA;
    const long pairB = pair0 + pB;
    const float* xpA = x + pairA * (TT * FF) + lane * 8;
    const float* xpB = x + pairB * (TT * FF) + lane * 8;

    float wqA[8], wqB[8];
    {
      v4f a0 = *(const v4f*)(wq_lds + pA * FF + lane * 8);
      v4f a1 = *(const v4f*)(wq_lds + pA * FF + lane * 8 + 4);
      v4f b0 = *(const v4f*)(wq_lds + pB * FF + lane * 8);
      v4f b1 = *(const v4f*)(wq_lds + pB * FF + lane * 8 + 4);
#pragma unroll
      for (int i = 0; i < 4; ++i) {
        wqA[i] = a0[i]; wqA[4 + i] = a1[i];
        wqB[i] = b0[i]; wqB[4 + i] = b1[i];
      }
    }

    float accA[8], accB[8];
#pragma unroll
    for (int i = 0; i < 8; ++i) { accA[i] = 0.f; accB[i] = 0.f; }
    float denA = 0.f, denB = 0.f;
    float mA = -INFINITY, mB = -INFINITY;

#pragma unroll 2
    for (int t = 0; t < TT; ++t) {
      // single-use stream: non-temporal so W/L2 isn't evicted
      v4f a0 = __builtin_nontemporal_load((const v4f*)(xpA + t * FF));
      v4f a1 = __builtin_nontemporal_load((const v4f*)(xpA + t * FF + 4));
      v4f b0 = __builtin_nontemporal_load((const v4f*)(xpB + t * FF));
      v4f b1 = __builtin_nontemporal_load((const v4f*)(xpB + t * FF + 4));
      float xa[8], xb[8];
#pragma unroll
      for (int i = 0; i < 4; ++i) {
        xa[i] = a0[i]; xa[4 + i] = a1[i];
        xb[i] = b0[i]; xb[4 + i] = b1[i];
      }

      float sA = 0.f, sB = 0.f;
#pragma unroll
      for (int i = 0; i < 8; ++i) {
        sA = fmaf(xa[i], wqA[i], sA);
        sB = fmaf(xb[i], wqB[i], sB);
      }
      sA = wave_reduce_add(sA);   // every lane: full score s_t (pair A)
      sB = wave_reduce_add(sB);   // every lane: full score s_t (pair B)

      // online softmax update, pair A
      {
        const float m2    = fmaxf(mA, sA);
        const float scale = __expf(mA - m2);   // exp(-inf)=0 on first step
        const float e     = __expf(sA - m2);
        denA = denA * scale + e;
#pragma unroll
        for (int i = 0; i < 8; ++i) accA[i] = fmaf(accA[i], scale, e * xa[i]);
        mA = m2;
      }
      // online softmax update, pair B
      {
        const float m2    = fmaxf(mB, sB);
        const float scale = __expf(mB - m2);
        const float e     = __expf(sB - m2);
        denB = denB * scale + e;
#pragma unroll
        for (int i = 0; i < 8; ++i) accB[i] = fmaf(accB[i], scale, e * xb[i]);
        mB = m2;
      }
    }

    const float invA = 1.0f / denA;
    const float invB = 1.0f / denB;
    v4f oA0, oA1, oB0, oB1;
#pragma unroll
    for (int i = 0; i < 4; ++i) {
      oA0[i] = accA[i] * invA; oA1[i] = accA[4 + i] * invA;
      oB0[i] = accB[i] * invB; oB1[i] = accB[4 + i] * invB;
    }
    float* opA = out + pairA * FF + lane * 8;
    float* opB = out + pairB * FF + lane * 8;
    __builtin_nontemporal_store(oA0, (v4f*)(opA));
    __builtin_nontemporal_store(oA1, (v4f*)(opA + 4));
    __builtin_nontemporal_store(oB0, (v4f*)(opB));
    __builtin_nontemporal_store(oB1, (v4f*)(opB + 4));
  }
}

extern "C" void kernel_launch(void* const* d_in, const int* in_sizes, int n_in,
                              void* d_out, int out_size, void* d_ws, size_t ws_size,
                              hipStream_t stream) {
  const float* x = (const float*)d_in[0];       // (16,512,64,256) f32
  const float* w = (const float*)d_in[1];       // (256,256) f32
  float* out = (float*)d_out;                   // (16,512,256) f32
  (void)in_sizes; (void)n_in; (void)out_size; (void)d_ws; (void)ws_size;

  const int blocks = (BB * SS) / GROUP;         // 512
  temporal_agg_kernel<<<blocks, THREADS, 0, stream>>>(x, w, out);
}